// NasBenchGINPredictorAgentCELU_24970939859297
// MI455X (gfx1250) — compile-verified
//
#include <hip/hip_runtime.h>
#include <math.h>

// ---------------------------------------------------------------------------
// GIN predictor for MI455X (gfx1250, wave32).
// Edge aggregation: lane-per-feature global_atomic_add_f32 (memory-bound part;
// h and agg both fit in the 192MB L2, so gathers/atomics stay on-chip).
// Node MLPs: V_WMMA_F32_16X16X4_F32, 16-node tiles per wave, fp32 end to end.
// ---------------------------------------------------------------------------

typedef float v2f __attribute__((ext_vector_type(2)));
typedef float v8f __attribute__((ext_vector_type(8)));

#define WPB 4  // waves per block in MLP kernels (128 threads)

__device__ __forceinline__ v8f wmma4(v2f a, v2f b, v8f c) {
  // D(16x16 f32) = A(16x4 f32) * B(4x16 f32) + C
  return __builtin_amdgcn_wmma_f32_16x16x4_f32(false, a, false, b, (short)0, c,
                                               false, false);
}

// Exact branchless CELU(alpha=1): for v>0 the exp term is exp(0)-1 == 0.
__device__ __forceinline__ float celu1(float v) {
  return fmaxf(v, 0.f) + (expf(fminf(v, 0.f)) - 1.f);
}

// [16 x 32] (LDS, padded rows) @ [32 x 32] + bias -> two 16x16 accumulators.
// A lane layout per ISA: lanes 0-15 hold K = kb+0/+1, lanes 16-31 K = kb+2/+3.
__device__ __forceinline__ void gemm_tile(const float (*A)[33],
                                          const float* __restrict__ W,
                                          const float* __restrict__ bias,
                                          int lane, v8f out[2]) {
  const int m = lane & 15;
  const int khalf = (lane >> 4) * 2;
#pragma unroll
  for (int nt = 0; nt < 2; ++nt) {
    const int n = (lane & 15) + nt * 16;
    const float bv = bias[n];
    v8f c = {bv, bv, bv, bv, bv, bv, bv, bv};
#pragma unroll
    for (int kb = 0; kb < 32; kb += 4) {
      const int k = kb + khalf;
      v2f a;
      a.x = A[m][k];
      a.y = A[m][k + 1];
      v2f b;
      b.x = W[k * 32 + n];
      b.y = W[(k + 1) * 32 + n];
      c = wmma4(a, b, c);
    }
    out[nt] = c;
  }
}

// Second GEMM + CELU + store pre-BN activations + sum/sumsq accumulation.
__device__ __forceinline__ void mlp_tail(const float (*tR)[33],
                                         const float* __restrict__ W2,
                                         const float* __restrict__ b2,
                                         float* __restrict__ outp,
                                         float* __restrict__ gsum,
                                         float* __restrict__ gsumsq,
                                         float* bsum, float* bsq,
                                         long long node0, int nN, int lane) {
  v8f d[2];
  gemm_tile(tR, W2, b2, lane, d);
  const int half = lane >> 4;
  float s1[2] = {0.f, 0.f}, s2[2] = {0.f, 0.f};
#pragma unroll
  for (int nt = 0; nt < 2; ++nt) {
    const int col = (lane & 15) + nt * 16;
#pragma unroll
    for (int i = 0; i < 8; ++i) {
      const int row = i + half * 8;
      const long long nrow = node0 + row;
      float v = celu1(d[nt][i]);
      if (nrow < nN) {
        outp[(size_t)nrow * 32 + col] = v;
        s1[nt] += v;
        s2[nt] += v * v;
      }
    }
  }
  atomicAdd(&bsum[lane & 15], s1[0]);  // ds_add_f32
  atomicAdd(&bsq[lane & 15], s2[0]);
  atomicAdd(&bsum[(lane & 15) + 16], s1[1]);
  atomicAdd(&bsq[(lane & 15) + 16], s2[1]);
  __syncthreads();
  if (threadIdx.x < 32) {
    unsafeAtomicAdd(&gsum[threadIdx.x], bsum[threadIdx.x]);
    unsafeAtomicAdd(&gsumsq[threadIdx.x], bsq[threadIdx.x]);
  }
}

// ---------------- zero / init ----------------
__global__ void zero_f32_kernel(float* __restrict__ p, long long n) {
  long long stride = (long long)gridDim.x * blockDim.x;
  for (long long i = (long long)blockIdx.x * blockDim.x + threadIdx.x; i < n;
       i += stride)
    p[i] = 0.f;
}

__global__ void zero_u32_kernel(unsigned* __restrict__ p, long long n) {
  long long stride = (long long)gridDim.x * blockDim.x;
  for (long long i = (long long)blockIdx.x * blockDim.x + threadIdx.x; i < n;
       i += stride)
    p[i] = 0u;
}

// ---------------- edge aggregation ----------------
// Layer 1: 6-dim features straight from x.
__global__ void agg6_kernel(const int* __restrict__ ei,
                            const float* __restrict__ x,
                            float* __restrict__ agg6, long long E) {
  const long long tot = E * 6;
  const long long stride = (long long)gridDim.x * blockDim.x;
  for (long long i = (long long)blockIdx.x * blockDim.x + threadIdx.x; i < tot;
       i += stride) {
    const long long e = i / 6;
    const int f = (int)(i - e * 6);
    const int s = ei[e];
    const int d = ei[E + e];
    unsafeAtomicAdd(&agg6[(size_t)d * 6 + f], x[(size_t)s * 6 + f]);
  }
}

// Layers 2/3: wave per edge, lane = feature (coalesced 128B gather + scatter).
__global__ void agg32_kernel(const int* __restrict__ ei,
                             const float* __restrict__ h,
                             float* __restrict__ agg, long long E) {
  const int lane = threadIdx.x & 31;
  const long long w =
      (long long)blockIdx.x * (blockDim.x >> 5) + (threadIdx.x >> 5);
  const long long nw = (long long)gridDim.x * (blockDim.x >> 5);
  for (long long e = w; e < E; e += nw) {
    const int s = ei[e];
    const int d = ei[E + e];
    const float v = h[(size_t)s * 32 + lane];
    unsafeAtomicAdd(&agg[(size_t)d * 32 + lane], v);
  }
}

// ---------------- node MLPs ----------------
// Layer 1: (x+agg)[N x 6] @ W1a[6x32] (scalar, K=6) -> relu -> WMMA 32x32.
__global__ void __launch_bounds__(128) mlp6_kernel(
    const float* __restrict__ x, const float* __restrict__ agg6,
    float* __restrict__ hp, const float* __restrict__ W1,
    const float* __restrict__ b1, const float* __restrict__ W2,
    const float* __restrict__ b2, float* __restrict__ gsum,
    float* __restrict__ gsumsq, int nN) {
  __shared__ float tR[WPB][16][33];
  __shared__ float bsum[32], bsq[32];
  const int lane = threadIdx.x & 31;
  const int wave = threadIdx.x >> 5;
  if (threadIdx.x < 32) {
    bsum[threadIdx.x] = 0.f;
    bsq[threadIdx.x] = 0.f;
  }
  __syncthreads();

  const long long node0 = (long long)(blockIdx.x * WPB + wave) * 16;
#pragma unroll 1
  for (int m = 0; m < 16; ++m) {
    const long long row = node0 + m;
    float acc = b1[lane];
    if (row < nN) {
#pragma unroll
      for (int k = 0; k < 6; ++k) {
        const float t = x[(size_t)row * 6 + k] + agg6[(size_t)row * 6 + k];
        acc += t * W1[k * 32 + lane];
      }
    }
    tR[wave][m][lane] = fmaxf(acc, 0.f);
  }
  mlp_tail(tR[wave], W2, b2, hp, gsum, gsumsq, bsum, bsq, node0, nN, lane);
}

// Layers 2/3: WMMA for both 32x32 GEMMs. hp written in place over agg (each
// wave only touches its own 16 rows: stage-read first, write after).
__global__ void __launch_bounds__(128) mlp32_kernel(
    const float* __restrict__ hin, float* __restrict__ aggp,
    const float* __restrict__ W1, const float* __restrict__ b1,
    const float* __restrict__ W2, const float* __restrict__ b2,
    float* __restrict__ gsum, float* __restrict__ gsumsq, int nN) {
  __shared__ float tA[WPB][16][33];
  __shared__ float tR[WPB][16][33];
  __shared__ float bsum[32], bsq[32];
  const int lane = threadIdx.x & 31;
  const int wave = threadIdx.x >> 5;
  if (threadIdx.x < 32) {
    bsum[threadIdx.x] = 0.f;
    bsq[threadIdx.x] = 0.f;
  }
  __syncthreads();

  const long long node0 = (long long)(blockIdx.x * WPB + wave) * 16;
#pragma unroll 1
  for (int m = 0; m < 16; ++m) {
    const long long row = node0 + m;
    float v = 0.f;
    if (row < nN) {
      const size_t off = (size_t)row * 32 + lane;
      v = hin[off] + aggp[off];
    }
    tA[wave][m][lane] = v;
  }
  v8f r1[2];
  gemm_tile(tA[wave], W1, b1, lane, r1);
  const int half = lane >> 4;
#pragma unroll
  for (int nt = 0; nt < 2; ++nt) {
    const int col = (lane & 15) + nt * 16;
#pragma unroll
    for (int i = 0; i < 8; ++i)
      tR[wave][i + half * 8][col] = fmaxf(r1[nt][i], 0.f);
  }
  mlp_tail(tR[wave], W2, b2, aggp, gsum, gsumsq, bsum, bsq, node0, nN, lane);
}

// ---------------- BatchNorm ----------------
__global__ void bn_final_kernel(float* __restrict__ gsum,
                                float* __restrict__ gsumsq,
                                const float* __restrict__ g,
                                const float* __restrict__ be,
                                float* __restrict__ scale,
                                float* __restrict__ shift, float invN) {
  const int t = threadIdx.x;  // 32 threads
  const float mu = gsum[t] * invN;
  const float var = gsumsq[t] * invN - mu * mu;
  const float rs = rsqrtf(var + 1e-5f);
  const float sc = rs * g[t];
  scale[t] = sc;
  shift[t] = be[t] - mu * sc;
  gsum[t] = 0.f;  // reset for next layer (deterministic within one launch)
  gsumsq[t] = 0.f;
}

__global__ void bn_apply_kernel(const float* __restrict__ src,
                                float* __restrict__ dst,
                                const float* __restrict__ scale,
                                const float* __restrict__ shift,
                                long long n32) {
  const long long stride = (long long)gridDim.x * blockDim.x;
  for (long long i = (long long)blockIdx.x * blockDim.x + threadIdx.x; i < n32;
       i += stride) {
    const int c = (int)(i & 31);
    dst[i] = src[i] * scale[c] + shift[c];
  }
}

// ---------------- global max pool ----------------
// batch is sorted; each wave scans a contiguous node chunk (lane = feature),
// carries a running max and flushes one atomicMax per graph-run (~10x fewer
// atomics than per-node). Order-preserving float->u32 key transform; untouched
// graphs keep key 0, which decodes to NaN and is guarded to 0 in the head.
#define POOL_CHUNK 128

__device__ __forceinline__ unsigned pool_enc(float v) {
  const unsigned b = __float_as_uint(v);
  return (b & 0x80000000u) ? ~b : (b | 0x80000000u);
}

__global__ void pool_kernel(const float* __restrict__ h,
                            const int* __restrict__ batch,
                            unsigned* __restrict__ keys, int nN) {
  const int lane = threadIdx.x & 31;
  const long long w =
      (long long)blockIdx.x * (blockDim.x >> 5) + (threadIdx.x >> 5);
  const long long n0 = w * POOL_CHUNK;
  if (n0 >= nN) return;
  const long long n1 = (n0 + POOL_CHUNK < nN) ? n0 + POOL_CHUNK : nN;
  int curg = -1;
  float run = 0.f;
#pragma unroll 1
  for (long long n = n0; n < n1; ++n) {
    const int g = batch[n];  // broadcast load (sorted)
    const float v = h[(size_t)n * 32 + lane];
    if (g != curg) {
      if (curg >= 0) atomicMax(&keys[(size_t)curg * 32 + lane], pool_enc(run));
      curg = g;
      run = v;
    } else {
      run = fmaxf(run, v);
    }
  }
  if (curg >= 0) atomicMax(&keys[(size_t)curg * 32 + lane], pool_enc(run));
}

// ---------------- head: celu(emb@Wb+bb) @ Wm + bm -> sigmoid ----------------
__global__ void head_kernel(const unsigned* __restrict__ keys,
                            const float* __restrict__ Wb,
                            const float* __restrict__ bb,
                            const float* __restrict__ Wm,
                            const float* __restrict__ bm,
                            float* __restrict__ out, int G) {
  const int g = blockIdx.x * blockDim.x + threadIdx.x;
  if (g >= G) return;
  float e[32];
#pragma unroll
  for (int k = 0; k < 32; ++k) {
    const unsigned key = keys[(size_t)g * 32 + k];
    const unsigned b = (key & 0x80000000u) ? (key & 0x7fffffffu) : ~key;
    const float v = __uint_as_float(b);
    e[k] = __builtin_isfinite(v) ? v : 0.f;  // untouched graphs decode to NaN
  }
  float acc2 = bm[0];
#pragma unroll 1
  for (int j = 0; j < 16; ++j) {
    float acc = bb[j];
#pragma unroll
    for (int k = 0; k < 32; ++k) acc += e[k] * Wb[k * 16 + j];
    acc2 += celu1(acc) * Wm[j];
  }
  out[g] = 1.f / (1.f + expf(-acc2));
}

// ---------------------------------------------------------------------------
extern "C" void kernel_launch(void* const* d_in, const int* in_sizes, int n_in,
                              void* d_out, int out_size, void* d_ws,
                              size_t ws_size, hipStream_t stream) {
  const float* x = (const float*)d_in[0];
  const int* ei = (const int*)d_in[1];
  const int* batch = (const int*)d_in[2];
  const float* W1a = (const float*)d_in[3];
  const float* b1a = (const float*)d_in[4];
  const float* W1b = (const float*)d_in[5];
  const float* b1b = (const float*)d_in[6];
  const float* g1 = (const float*)d_in[7];
  const float* be1 = (const float*)d_in[8];
  const float* W2a = (const float*)d_in[9];
  const float* b2a = (const float*)d_in[10];
  const float* W2b = (const float*)d_in[11];
  const float* b2b = (const float*)d_in[12];
  const float* g2 = (const float*)d_in[13];
  const float* be2 = (const float*)d_in[14];
  const float* W3a = (const float*)d_in[15];
  const float* b3a = (const float*)d_in[16];
  const float* W3b = (const float*)d_in[17];
  const float* b3b = (const float*)d_in[18];
  const float* g3 = (const float*)d_in[19];
  const float* be3 = (const float*)d_in[20];
  const float* Wb = (const float*)d_in[21];
  const float* bb = (const float*)d_in[22];
  const float* Wm = (const float*)d_in[23];
  const float* bm = (const float*)d_in[24];

  const int nN = in_sizes[2];           // 500000
  const long long E = in_sizes[1] / 2;  // 8000000
  const int G = out_size;               // 50000
  const float invN = 1.f / (float)nN;

  float* ws = (float*)d_ws;
  const size_t N32 = (size_t)nN * 32;
  float* buf_h = ws;          // N x 32 current activations
  float* buf_agg = ws + N32;  // N x 32 aggregation / pre-BN scratch
  unsigned* keys = (unsigned*)(ws + 2 * N32);  // G x 32 pooling keys
  float* stats = ws + 2 * N32 + (size_t)G * 32;
  float* gsum = stats;
  float* gsumsq = stats + 32;
  float* scale = stats + 64;
  float* shift = stats + 96;

  const int tiles = (nN + 15) / 16;
  const int mlpBlocks = (tiles + WPB - 1) / WPB;
  const long long n32 = (long long)nN * 32;
  const int ZB = 4096, EB = 8192;
  float* out = (float*)d_out;

  // ---- layer 1 (fin = 6) ----
  zero_f32_kernel<<<1, 64, 0, stream>>>(gsum, 64);  // gsum+gsumsq contiguous
  zero_f32_kernel<<<ZB, 256, 0, stream>>>(buf_agg, (long long)nN * 6);
  agg6_kernel<<<EB, 256, 0, stream>>>(ei, x, buf_agg, E);
  mlp6_kernel<<<mlpBlocks, 128, 0, stream>>>(x, buf_agg, buf_h, W1a, b1a, W1b,
                                             b1b, gsum, gsumsq, nN);
  bn_final_kernel<<<1, 32, 0, stream>>>(gsum, gsumsq, g1, be1, scale, shift,
                                        invN);
  bn_apply_kernel<<<ZB, 256, 0, stream>>>(buf_h, buf_h, scale, shift, n32);

  // ---- layer 2 ----
  zero_f32_kernel<<<ZB, 256, 0, stream>>>(buf_agg, n32);
  agg32_kernel<<<EB, 256, 0, stream>>>(ei, buf_h, buf_agg, E);
  mlp32_kernel<<<mlpBlocks, 128, 0, stream>>>(buf_h, buf_agg, W2a, b2a, W2b,
                                              b2b, gsum, gsumsq, nN);
  bn_final_kernel<<<1, 32, 0, stream>>>(gsum, gsumsq, g2, be2, scale, shift,
                                        invN);
  bn_apply_kernel<<<ZB, 256, 0, stream>>>(buf_agg, buf_h, scale, shift, n32);

  // ---- layer 3 ----
  zero_f32_kernel<<<ZB, 256, 0, stream>>>(buf_agg, n32);
  agg32_kernel<<<EB, 256, 0, stream>>>(ei, buf_h, buf_agg, E);
  mlp32_kernel<<<mlpBlocks, 128, 0, stream>>>(buf_h, buf_agg, W3a, b3a, W3b,
                                              b3b, gsum, gsumsq, nN);
  bn_final_kernel<<<1, 32, 0, stream>>>(gsum, gsumsq, g3, be3, scale, shift,
                                        invN);
  bn_apply_kernel<<<ZB, 256, 0, stream>>>(buf_agg, buf_h, scale, shift, n32);

  // ---- pool + head ----
  zero_u32_kernel<<<ZB, 256, 0, stream>>>(keys, (long long)G * 32);
  {
    const int poolWaves = (nN + POOL_CHUNK - 1) / POOL_CHUNK;
    const int poolBlocks = (poolWaves + 7) / 8;  // 8 waves per 256-thread block
    pool_kernel<<<poolBlocks, 256, 0, stream>>>(buf_h, batch, keys, nN);
  }
  head_kernel<<<(G + 255) / 256, 256, 0, stream>>>(keys, Wb, bb, Wm, bm, out,
                                                   G);
}